// GraphAttention_56796647522361
// MI455X (gfx1250) — compile-verified
//
#include <hip/hip_runtime.h>
#include <hip/hip_bf16.h>
#include <math.h>

typedef __attribute__((ext_vector_type(2))) float v2f;
typedef __attribute__((ext_vector_type(8))) float v8f;

#define F_IN   128
#define F_OUT  64
#define WAVES  8
#define THREADS 256
#define KSTEPS 32           // 128 / 4
#define NT     4            // 64 / 16
#define REC    66           // m, s, v[64]

// ---------------------------------------------------------------------------
// Pass 1: fused  X@kernel  ->  a = X_t@attn_neigh  ->  online softmax accum of
//         (m, s, v[64] = sum exp(a_r - m) * X_t[r][:])  per wave, merged per
//         block into a 66-float partial record.
// ---------------------------------------------------------------------------
__global__ __launch_bounds__(THREADS) void gat_pass1(
    const float* __restrict__ X, const float* __restrict__ kern,
    const float* __restrict__ attn_neigh, float* __restrict__ partials,
    int N, int nChunks)
{
  __shared__ float sB[KSTEPS * NT * 32 * 2];     // 8192 floats = 32 KB
  __shared__ float sMerge[WAVES * REC];

  const int tid = threadIdx.x;

  // Rearrange kernel (B matrix, 4x16 fragments) into per-fragment LDS layout:
  // sB[((ks*4+nt)*32 + lane)*2 + j] = kern[K][C] with
  //   K = ks*4 + (lane<16 ? 0 : 2) + j,  C = nt*16 + (lane&15)
  for (int idx = tid; idx < KSTEPS * NT * 32 * 2; idx += THREADS) {
    int j     = idx & 1;
    int l     = (idx >> 1) & 31;
    int fi    = idx >> 6;
    int kstep = fi >> 2;
    int nt    = fi & 3;
    int K = kstep * 4 + ((l < 16) ? 0 : 2) + j;
    int C = nt * 16 + (l & 15);
    sB[idx] = kern[K * F_OUT + C];
  }
  __syncthreads();

  const int  lane = tid & 31;
  const int  wave = tid >> 5;
  const int  lm   = lane & 15;
  const bool hi   = lane >= 16;

  // attn_neigh weight for this lane's 4 columns (one per n-tile)
  const float wv0 = attn_neigh[0 * 16 + lm];
  const float wv1 = attn_neigh[1 * 16 + lm];
  const float wv2 = attn_neigh[2 * 16 + lm];
  const float wv3 = attn_neigh[3 * 16 + lm];

  float m = -1e30f, s = 0.f;
  float vacc0 = 0.f, vacc1 = 0.f, vacc2 = 0.f, vacc3 = 0.f;

  const int gwave  = blockIdx.x * WAVES + wave;
  const int nwaves = gridDim.x * WAVES;
  const int rowStride = nwaves * 16;            // rows between my chunks

  for (int chunk = gwave; chunk < nChunks; chunk += nwaves) {
    const int rowBase = chunk * 16;

    // Prefetch next chunk's A tile (contiguous 16 rows = 8 KB = 64 lines) so
    // HBM fetch of chunk i+1 overlaps WMMA + softmax epilogue of chunk i.
    // global_prefetch_b8: one line per active lane, no LOADcnt cost.
    {
      const int nextBase = rowBase + rowStride;
      if (nextBase + 16 <= N) {                 // skip tail tile: stay mapped
        const float* pf = X + (size_t)nextBase * F_IN;
        __builtin_prefetch(pf + (size_t)lane * 32, 0, 3);        // lines 0..31
        __builtin_prefetch(pf + (size_t)(lane + 32) * 32, 0, 3); // lines 32..63
      }
    }

    int arow = rowBase + lm;                 // A-fragment row for this lane
    if (arow > N - 1) arow = N - 1;          // clamp: never read OOB
    const float* Abase = X + (size_t)arow * F_IN + (hi ? 2 : 0);

    v8f acc[NT] = {};                        // X_t tile, 16x64, fp32
    #pragma unroll
    for (int ks = 0; ks < KSTEPS; ++ks) {
      v2f a = *(const v2f*)(Abase + ks * 4);
      #pragma unroll
      for (int nt = 0; nt < NT; ++nt) {
        v2f b = *(const v2f*)(&sB[(ks * NT + nt) * 64 + lane * 2]);
        acc[nt] = __builtin_amdgcn_wmma_f32_16x16x4_f32(
            false, a, false, b, (short)0, acc[nt], false, false);
      }
    }

    // logits: pa[j] = partial of a[row] over this lane's 4 columns
    float pa[8];
    #pragma unroll
    for (int j = 0; j < 8; ++j)
      pa[j] = acc[0][j] * wv0 + acc[1][j] * wv1 + acc[2][j] * wv2 + acc[3][j] * wv3;
    // reduce over the 16 lanes that hold the same rows
    #pragma unroll
    for (int off = 1; off <= 8; off <<= 1) {
      #pragma unroll
      for (int j = 0; j < 8; ++j)
        pa[j] += __shfl_xor(pa[j], off, 32);
    }
    // pa[j] now = logit of row (hi?8:0)+j, replicated within the half-wave

    float cmax = pa[0];
    #pragma unroll
    for (int j = 1; j < 8; ++j) cmax = fmaxf(cmax, pa[j]);
    cmax = fmaxf(cmax, __shfl_xor(cmax, 16, 32));   // full 16-row chunk max

    const float mn = fmaxf(m, cmax);
    const float sc = __expf(m - mn);
    s *= sc; vacc0 *= sc; vacc1 *= sc; vacc2 *= sc; vacc3 *= sc;

    const int rb = rowBase + (hi ? 8 : 0);
    #pragma unroll
    for (int j = 0; j < 8; ++j) {
      const float e = ((rb + j) < N) ? __expf(pa[j] - mn) : 0.f;
      s += e;
      vacc0 += e * acc[0][j];
      vacc1 += e * acc[1][j];
      vacc2 += e * acc[2][j];
      vacc3 += e * acc[3][j];
    }
    m = mn;
  }

  // fold the two half-wave row groups together
  s     += __shfl_xor(s,     16, 32);
  vacc0 += __shfl_xor(vacc0, 16, 32);
  vacc1 += __shfl_xor(vacc1, 16, 32);
  vacc2 += __shfl_xor(vacc2, 16, 32);
  vacc3 += __shfl_xor(vacc3, 16, 32);

  float* rec = &sMerge[wave * REC];
  if (lane == 0) { rec[0] = m; rec[1] = s; }
  if (lane < 16) {
    rec[2 + 0 * 16 + lm] = vacc0;
    rec[2 + 1 * 16 + lm] = vacc1;
    rec[2 + 2 * 16 + lm] = vacc2;
    rec[2 + 3 * 16 + lm] = vacc3;
  }
  __syncthreads();

  // merge 8 wave records -> one block record
  if (tid < F_OUT) {
    float mb = sMerge[0];
    #pragma unroll
    for (int w = 1; w < WAVES; ++w) mb = fmaxf(mb, sMerge[w * REC]);
    float vb = 0.f, sb = 0.f;
    #pragma unroll
    for (int w = 0; w < WAVES; ++w) {
      const float e = __expf(sMerge[w * REC] - mb);
      vb += sMerge[w * REC + 2 + tid] * e;
      sb += sMerge[w * REC + 1] * e;
    }
    float* outp = &partials[(size_t)blockIdx.x * REC];
    if (tid == 0) { outp[0] = mb; outp[1] = sb; }
    outp[2 + tid] = vb;
  }
}

// ---------------------------------------------------------------------------
// Pass 2: combine block partials, add 0.25 * (x @ kernel), write 64 outputs.
// ---------------------------------------------------------------------------
__global__ __launch_bounds__(256) void gat_finalize(
    const float* __restrict__ partials, int nb,
    const float* __restrict__ x, const float* __restrict__ kern,
    float* __restrict__ out)
{
  __shared__ float red[256];
  const int tid = threadIdx.x;

  float lm = -1e30f;
  for (int i = tid; i < nb; i += 256) lm = fmaxf(lm, partials[(size_t)i * REC]);
  red[tid] = lm; __syncthreads();
  for (int off = 128; off > 0; off >>= 1) {
    if (tid < off) red[tid] = fmaxf(red[tid], red[tid + off]);
    __syncthreads();
  }
  const float M = red[0];
  __syncthreads();

  float ls = 0.f;
  for (int i = tid; i < nb; i += 256)
    ls += partials[(size_t)i * REC + 1] * __expf(partials[(size_t)i * REC] - M);
  red[tid] = ls; __syncthreads();
  for (int off = 128; off > 0; off >>= 1) {
    if (tid < off) red[tid] += red[tid + off];
    __syncthreads();
  }
  const float S = red[0];

  if (tid < F_OUT) {
    float v = 0.f;
    for (int i = 0; i < nb; ++i)
      v += partials[(size_t)i * REC + 2 + tid] * __expf(partials[(size_t)i * REC] - M);
    float xt = 0.f;
    #pragma unroll 8
    for (int k = 0; k < F_IN; ++k) xt += x[k] * kern[k * F_OUT + tid];
    out[tid] = 0.75f * (v / S) + 0.25f * xt;
  }
}

// ---------------------------------------------------------------------------
extern "C" void kernel_launch(void* const* d_in, const int* in_sizes, int n_in,
                              void* d_out, int out_size, void* d_ws, size_t ws_size,
                              hipStream_t stream) {
  const float* x          = (const float*)d_in[0];
  const float* X          = (const float*)d_in[1];
  const float* kern       = (const float*)d_in[2];
  // d_in[3] = attn_self: constant shift of all logits -> cancels in softmax
  const float* attn_neigh = (const float*)d_in[4];
  float* out = (float*)d_out;

  const int N       = in_sizes[1] / F_IN;
  const int nChunks = (N + 15) / 16;

  int nb = 512;                                   // block partials
  const size_t need = (size_t)nb * REC * sizeof(float);
  if (ws_size < need) {
    nb = (int)(ws_size / (REC * sizeof(float)));
    if (nb < 1) nb = 1;
  }
  const int maxUseful = (nChunks + WAVES - 1) / WAVES;
  if (nb > maxUseful) nb = maxUseful;
  if (nb < 1) nb = 1;

  float* partials = (float*)d_ws;
  gat_pass1<<<nb, THREADS, 0, stream>>>(X, kern, attn_neigh, partials, N, nChunks);
  gat_finalize<<<1, 256, 0, stream>>>(partials, nb, x, kern, out);
}